// ModelNew_29240137351231
// MI455X (gfx1250) — compile-verified
//
#include <hip/hip_runtime.h>
#include <math.h>

// Retention (linear attention with per-head exponential decay), chunkwise form.
// out[i] = scale * sum_{j<=i} gamma^(i-j) (q_i . k_j) v_j
// Chunk C=64: intra-chunk via masked QK^T, cross-chunk via recurrent state
//   R_{t+1} = gamma^C R_t + A_t,  A_t = sum_li gamma^(C-li) k_li v_li^T
// All 64x64x64 GEMMs use V_WMMA_F32_16X16X4_F32 (full fp32 matrix pipe).
// Global->LDS staging uses GLOBAL_LOAD_ASYNC_TO_LDS_B128 (ASYNCcnt path).

typedef float v2f __attribute__((ext_vector_type(2)));
typedef float v8f __attribute__((ext_vector_type(8)));
typedef int   i32x4 __attribute__((vector_size(16)));   // matches builtin pointee type

#define WMMA_F32(a, b, c) \
  __builtin_amdgcn_wmma_f32_16x16x4_f32(false, (a), false, (b), (short)0, (c), false, false)

#if defined(__has_builtin)
#if __has_builtin(__builtin_amdgcn_global_load_async_to_lds_b128) && \
    __has_builtin(__builtin_amdgcn_s_wait_asynccnt)
#define USE_ASYNC 1
#endif
#endif
#ifndef USE_ASYNC
#define USE_ASYNC 0
#endif

constexpr int Bsz   = 2;
constexpr int H     = 16;
constexpr int S     = 2048;
constexpr int D     = 64;
constexpr int CHUNK = 64;
constexpr int NC    = S / CHUNK;   // 32 chunks per sequence
constexpr int LDST  = 68;          // LDS row stride (floats): 272B rows, 16B aligned,
                                   // pad keeps transposed col reads bank-conflict-free

// 16-byte global -> LDS copy (per lane). Async engine path if available.
__device__ __forceinline__ void cp16(const float* g, float* l) {
#if USE_ASYNC
  __builtin_amdgcn_global_load_async_to_lds_b128(
      (__attribute__((address_space(1))) i32x4*)(g),
      (__attribute__((address_space(3))) i32x4*)(l), 0, 0);
#else
  *(float4*)l = *(const float4*)g;
#endif
}

__device__ __forceinline__ void cp16_wait() {
#if USE_ASYNC
  __builtin_amdgcn_s_wait_asynccnt(0);
#endif
}

// ---------------------------------------------------------------------------
// Kernel 1: per-chunk KV summary  A_t[dk][dv] = sum_li gamma^(C-li) K[li][dk] V[li][dv]
// One block (8 waves) per (b,h,chunk). 64x64 output, K-dim = 64 (li).
// V rows scaled in LDS after async staging, keeping the WMMA loop pure.
// ---------------------------------------------------------------------------
__global__ __launch_bounds__(256) void retn_chunk_kv(const float* __restrict__ k,
                                                     const float* __restrict__ v,
                                                     float* __restrict__ kvsum) {
  __shared__ __align__(16) float Ks[CHUNK][LDST];
  __shared__ __align__(16) float Vs[CHUNK][LDST];

  const int t   = blockIdx.x % NC;
  const int bh  = blockIdx.x / NC;
  const int h   = bh % H;
  const float lg = log2f(1.0f - exp2f(-5.0f - (float)h));   // log2(gamma_h)
  const int tid = threadIdx.x;
  const size_t base = ((size_t)bh * S + (size_t)t * CHUNK) * D;  // chunk = 4096 contiguous floats

  for (int i = 0; i < 4; ++i) {
    const int f4  = tid + 256 * i;       // 0..1023 16B packets
    const int row = f4 >> 4;
    const int col = (f4 & 15) * 4;
    cp16(k + base + 4 * (size_t)f4, &Ks[row][col]);
    cp16(v + base + 4 * (size_t)f4, &Vs[row][col]);
  }
  cp16_wait();
  __syncthreads();                       // staged data visible to all waves

  // Scale V row li by gamma^(C-li), in place (outside the WMMA loop).
  for (int i = 0; i < 4; ++i) {
    const int f4  = tid + 256 * i;
    const int row = f4 >> 4;
    const int col = (f4 & 15) * 4;
    const float w = exp2f((float)(CHUNK - row) * lg);
    Vs[row][col + 0] *= w; Vs[row][col + 1] *= w;
    Vs[row][col + 2] *= w; Vs[row][col + 3] *= w;
  }
  __syncthreads();

  const int lane = tid & 31;
  const int wv   = tid >> 5;
  const int ln   = lane & 15;
  const int half = lane >> 4;
  float* outp = kvsum + (size_t)blockIdx.x * (D * D);

  for (int i = 0; i < 2; ++i) {                // 16 tiles / 8 waves = 2 tiles each
    const int tile = wv + 8 * i;
    const int tm = (tile >> 2) * 16;
    const int tn = (tile & 3) * 16;
    v8f c = {};
    for (int kk = 0; kk < CHUNK; kk += 4) {
      const int k0 = kk + 2 * half;            // lanes 16-31 handle K+2,K+3
      v2f a, b;
      a.x = Ks[k0][tm + ln];   a.y = Ks[k0 + 1][tm + ln];   // A = K^T (A[dk][li])
      b.x = Vs[k0][tn + ln];   b.y = Vs[k0 + 1][tn + ln];   // B = diag(w) V
      c = WMMA_F32(a, b, c);
    }
    for (int r = 0; r < 8; ++r) {
      const int row = tm + r + 8 * half;       // C/D layout: VGPR r -> M=r (lo half) / r+8
      outp[row * D + tn + ln] = c[r];
    }
  }
}

// ---------------------------------------------------------------------------
// Kernel 2: sequential scan over chunks, in place.
// Slot t gets R_t (state BEFORE chunk t);  R_{t+1} = gamma^C R_t + A_t.
// One block per (b,h); each thread owns 16 floats of the 64x64 state.
// ---------------------------------------------------------------------------
__global__ __launch_bounds__(256) void retn_scan(float* __restrict__ kvsum) {
  const int bh = blockIdx.x;
  const int h  = bh % H;
  const float lg = log2f(1.0f - exp2f(-5.0f - (float)h));
  const float gC = exp2f((float)CHUNK * lg);   // gamma^C
  const int tid = threadIdx.x;

  float4 R[4] = {};                            // R_0 = 0
  float* base = kvsum + (size_t)bh * NC * (D * D);
  for (int t = 0; t < NC; ++t) {
    float4* slot = (float4*)(base + (size_t)t * (D * D));
    for (int i = 0; i < 4; ++i) {
      const float4 a = slot[tid + 256 * i];    // A_t
      slot[tid + 256 * i] = R[i];              // overwrite with R_t
      R[i].x = gC * R[i].x + a.x;
      R[i].y = gC * R[i].y + a.y;
      R[i].z = gC * R[i].z + a.z;
      R[i].w = gC * R[i].w + a.w;
    }
  }
}

// ---------------------------------------------------------------------------
// Kernel 3: per-chunk output.
//   P[i][j]   = scale * gamma^(i-j) * (Q K^T)[i][j]  (i>=j, else 0)
//   out[i][:] = (P V)[i][:] + scale * gamma^i * (Q R_t)[i][:]
// One block (8 waves) per (b,h,chunk).
// ---------------------------------------------------------------------------
__global__ __launch_bounds__(256) void retn_out(const float* __restrict__ q,
                                                const float* __restrict__ k,
                                                const float* __restrict__ v,
                                                const float* __restrict__ state,
                                                float* __restrict__ out) {
  __shared__ __align__(16) float Qs[CHUNK][LDST];
  __shared__ __align__(16) float Bs[CHUNK][LDST];   // holds K, then V, then R
  __shared__ __align__(16) float Ps[CHUNK][LDST];

  const int t   = blockIdx.x % NC;
  const int bh  = blockIdx.x / NC;
  const int h   = bh % H;
  const float lg    = log2f(1.0f - exp2f(-5.0f - (float)h));
  const float scale = 0.125f;          // 1/sqrt(64)
  const int tid = threadIdx.x;
  const size_t base = ((size_t)bh * S + (size_t)t * CHUNK) * D;

  // Stage Q and K.
  for (int i = 0; i < 4; ++i) {
    const int f4  = tid + 256 * i;
    const int row = f4 >> 4;
    const int col = (f4 & 15) * 4;
    cp16(q + base + 4 * (size_t)f4, &Qs[row][col]);
    cp16(k + base + 4 * (size_t)f4, &Bs[row][col]);
  }
  cp16_wait();
  __syncthreads();

  const int lane = tid & 31;
  const int wv   = tid >> 5;
  const int ln   = lane & 15;
  const int half = lane >> 4;

  // P = (Q K^T) with decay mask + scale, into LDS.
  for (int i = 0; i < 2; ++i) {
    const int tile = wv + 8 * i;
    const int tm = (tile >> 2) * 16;
    const int tn = (tile & 3) * 16;
    v8f c = {};
    for (int kk = 0; kk < D; kk += 4) {
      const int k0 = kk + 2 * half;
      v2f a, b;
      a.x = Qs[tm + ln][k0];  a.y = Qs[tm + ln][k0 + 1];
      b.x = Bs[tn + ln][k0];  b.y = Bs[tn + ln][k0 + 1];   // B = K^T: B[d][j] = K[j][d]
      c = WMMA_F32(a, b, c);
    }
    for (int r = 0; r < 8; ++r) {
      const int row = tm + r + 8 * half;
      const int col = tn + ln;
      const float m = (row >= col) ? scale * exp2f((float)(row - col) * lg) : 0.0f;
      Ps[row][col] = c[r] * m;
    }
  }
  __syncthreads();                     // P done; K reads done -> Bs reusable

  // Stage V.
  for (int i = 0; i < 4; ++i) {
    const int f4  = tid + 256 * i;
    const int row = f4 >> 4;
    const int col = (f4 & 15) * 4;
    cp16(v + base + 4 * (size_t)f4, &Bs[row][col]);
  }
  cp16_wait();
  __syncthreads();

  // O1 = P * V  (keep in registers).
  v8f acc[2];
  for (int i = 0; i < 2; ++i) {
    const int tile = wv + 8 * i;
    const int tm = (tile >> 2) * 16;
    const int tn = (tile & 3) * 16;
    v8f c = {};
    for (int kk = 0; kk < CHUNK; kk += 4) {
      const int k0 = kk + 2 * half;
      v2f a, b;
      a.x = Ps[tm + ln][k0];  a.y = Ps[tm + ln][k0 + 1];
      b.x = Bs[k0][tn + ln];  b.y = Bs[k0 + 1][tn + ln];
      c = WMMA_F32(a, b, c);
    }
    acc[i] = c;
  }
  __syncthreads();                     // V reads done -> Bs reusable

  // Stage R_t (cross-chunk state).
  const float* Rp = state + (size_t)blockIdx.x * (D * D);
  for (int i = 0; i < 4; ++i) {
    const int f4  = tid + 256 * i;
    const int row = f4 >> 4;
    const int col = (f4 & 15) * 4;
    cp16(Rp + 4 * (size_t)f4, &Bs[row][col]);
  }
  cp16_wait();
  __syncthreads();

  // O2 = Q * R;  out = O1 + scale * gamma^i * O2.
  for (int i = 0; i < 2; ++i) {
    const int tile = wv + 8 * i;
    const int tm = (tile >> 2) * 16;
    const int tn = (tile & 3) * 16;
    v8f c = {};
    for (int kk = 0; kk < D; kk += 4) {
      const int k0 = kk + 2 * half;
      v2f a, b;
      a.x = Qs[tm + ln][k0];  a.y = Qs[tm + ln][k0 + 1];
      b.x = Bs[k0][tn + ln];  b.y = Bs[k0 + 1][tn + ln];
      c = WMMA_F32(a, b, c);
    }
    for (int r = 0; r < 8; ++r) {
      const int row = tm + r + 8 * half;
      const int col = tn + ln;
      const float rf = scale * exp2f((float)row * lg);
      out[base + (size_t)row * D + col] = acc[i][r] + rf * c[r];
    }
  }
}

// ---------------------------------------------------------------------------
extern "C" void kernel_launch(void* const* d_in, const int* in_sizes, int n_in,
                              void* d_out, int out_size, void* d_ws, size_t ws_size,
                              hipStream_t stream) {
  const float* q = (const float*)d_in[0];
  const float* k = (const float*)d_in[1];
  const float* v = (const float*)d_in[2];
  float* out   = (float*)d_out;
  float* kvsum = (float*)d_ws;   // needs B*H*NC*D*D*4 = 16 MiB of scratch

  dim3 blk(256);
  retn_chunk_kv<<<dim3(Bsz * H * NC), blk, 0, stream>>>(k, v, kvsum);
  retn_scan    <<<dim3(Bsz * H),      blk, 0, stream>>>(kvsum);
  retn_out     <<<dim3(Bsz * H * NC), blk, 0, stream>>>(q, k, v, kvsum, out);
}